// ScatterMultiHeadAttentionLayer_84559316123888
// MI455X (gfx1250) — compile-verified
//
#include <hip/hip_runtime.h>
#include <math.h>

// ---------------------------------------------------------------------------
// Problem constants (from the reference): N_Q=50000, N_K=500000, D=256, H=8
// ---------------------------------------------------------------------------
#define NQ 50000
#define NK 500000
#define DD 256
#define NH 8
#define HSZ 32                    // head size = D/H
#define RSQRT_HS 0.17677669529663687f  // 1/sqrt(32)

typedef __attribute__((ext_vector_type(2))) float v2f;
typedef __attribute__((ext_vector_type(4))) float v4f;
typedef __attribute__((ext_vector_type(8))) float v8f;

// ---------------------------------------------------------------------------
// Phase 0: zero the output accumulator (harness poisons d_out with 0xAA)
// ---------------------------------------------------------------------------
__global__ __launch_bounds__(256) void zero_kernel(float* __restrict__ out, int n4) {
    int i = blockIdx.x * blockDim.x + threadIdx.x;
    if (i < n4) {
        v4f z = {0.f, 0.f, 0.f, 0.f};
        ((v4f*)out)[i] = z;
    }
}

// ---------------------------------------------------------------------------
// Phase 1: q = GELU_exact(query @ Wq^T)    (50000x256) = (50000x256)x(256x256)
//
// fp32 WMMA 16x16x4 (exact fp32 multiply-accumulate on the matrix pipe).
// Block = 256 threads = 8 waves. Each wave computes one 16x16 tile of q.
// grid = (3125, 2): blockIdx.x = M-tile, blockIdx.y*8 + wave = N-tile (16 total).
//
// Fragment layouts (ISA 7.12.2, 32-bit A 16x4 / B 4x16 / C 16x16):
//   A: lane L, vgpr v ->  A[L%16][k0 + 2*(L/16) + v]
//   B: lane L, vgpr v ->  B[k0 + 2*(L/16) + v][L%16]   where B[k][n] = Wq[n][k]
//   C: lane L, vgpr r ->  C[r + 8*(L/16)][L%16]
// ---------------------------------------------------------------------------
#define APAD 4  // LDS row pad: bank = (260*row + k) % 64 = (4*row + k) % 64 -> conflict-free

__global__ __launch_bounds__(256) void gemm_gelu_kernel(const float* __restrict__ query,
                                                        const float* __restrict__ Wq,
                                                        float* __restrict__ q) {
    __shared__ float As[16][DD + APAD];  // 16.6 KB

    const int tid   = threadIdx.x;
    const int mtile = blockIdx.x;              // 0..3124 (50000/16)
    const int wave  = tid >> 5;                // 0..7
    const int lane  = tid & 31;
    const int ntile = blockIdx.y * 8 + wave;   // 0..15

    // Stage the 16x256 A tile into (padded) LDS: 1024 float4s, 4 per thread.
    {
        const v4f* src = (const v4f*)(query + (size_t)mtile * 16 * DD);
#pragma unroll
        for (int it = 0; it < 4; ++it) {
            int idx4 = it * 256 + tid;         // 0..1023
            int row  = idx4 >> 6;              // /64 float4 per row
            int col4 = idx4 & 63;
            *(v4f*)(&As[row][col4 * 4]) = src[idx4];
        }
    }
    __syncthreads();

    const int mrow = lane & 15;
    const int half = lane >> 4;                // 0: lanes 0-15, 1: lanes 16-31
    const int ncol = ntile * 16 + (lane & 15);
    const float* __restrict__ Brow = Wq + (size_t)ncol * DD;  // Wq row = output col

    v8f acc = {};
#pragma unroll 8
    for (int k0 = 0; k0 < DD; k0 += 4) {
        const int kk = k0 + half * 2;
        v2f a = *(const v2f*)(&As[mrow][kk]);
        v2f b = *(const v2f*)(Brow + kk);
        acc = __builtin_amdgcn_wmma_f32_16x16x4_f32(false, a, false, b,
                                                    (short)0, acc, false, false);
    }

    // Exact GELU: 0.5*x*(1+erf(x/sqrt(2))), then coalesced store per C-row.
    const int colg = ntile * 16 + (lane & 15);
#pragma unroll
    for (int r = 0; r < 8; ++r) {
        const int rowg = mtile * 16 + r + half * 8;
        float x = acc[r];
        float g = 0.5f * x * (1.0f + erff(x * 0.70710678118654752f));
        q[(size_t)rowg * DD + colg] = g;
    }
}

// ---------------------------------------------------------------------------
// Phase 2: per-edge attention + scatter-add.
//   For edge e: dot(q[qe[e]], key[e]) per head (32 dims), sigmoid(dot/sqrt(32)),
//   scale value[e], atomically accumulate into out[va[e]].
//
// 64 threads per edge, one float4 (4 dims) per thread; head = 8 lanes, so the
// head reduction is 3 shfl_xor steps that stay inside one wave32.
// key/value are stream-once (512 MB each) -> non-temporal loads, preserving
// L2 for the q gather table (51 MB) and the atomic destination (51 MB).
// ---------------------------------------------------------------------------
__global__ __launch_bounds__(256) void edge_kernel(const float* __restrict__ q,
                                                   const float* __restrict__ key,
                                                   const float* __restrict__ value,
                                                   const int* __restrict__ qidx,
                                                   const int* __restrict__ oidx,
                                                   float* __restrict__ out) {
    const int tid  = threadIdx.x;
    const int slot = tid >> 6;                 // 4 edges per block
    const int r    = tid & 63;                 // float4 slot within the 256-dim row
    const long e   = (long)blockIdx.x * 4 + slot;

    const int qi = qidx[e];
    const int oi = oidx[e];

    const v4f* __restrict__ kp = (const v4f*)key   + (size_t)e * 64 + r;
    const v4f* __restrict__ vp = (const v4f*)value + (size_t)e * 64 + r;
    const v4f* __restrict__ qp = (const v4f*)q     + (size_t)qi * 64 + r;

    v4f kv = __builtin_nontemporal_load(kp);
    v4f qv = *qp;

    float p = qv.x * kv.x + qv.y * kv.y + qv.z * kv.z + qv.w * kv.w;
    // reduce over the 8-lane head group (xor masks 1,2,4 stay within the group)
    p += __shfl_xor(p, 1, 32);
    p += __shfl_xor(p, 2, 32);
    p += __shfl_xor(p, 4, 32);

    const float w = 1.0f / (1.0f + __expf(-p * RSQRT_HS));

    v4f vv = __builtin_nontemporal_load(vp);

    float* op = out + (size_t)oi * DD + r * 4;
    atomicAdd(op + 0, w * vv.x);
    atomicAdd(op + 1, w * vv.y);
    atomicAdd(op + 2, w * vv.z);
    atomicAdd(op + 3, w * vv.w);
}

// ---------------------------------------------------------------------------
// Launcher
//   d_in: query, key, value, Wq, query_expand, value_aggregate, o_shape
//   d_ws: holds q = GELU(query @ Wq^T)  (50000*256 fp32 = 51.2 MB)
// ---------------------------------------------------------------------------
extern "C" void kernel_launch(void* const* d_in, const int* in_sizes, int n_in,
                              void* d_out, int out_size, void* d_ws, size_t ws_size,
                              hipStream_t stream) {
    (void)in_sizes; (void)n_in; (void)out_size; (void)ws_size;

    const float* query = (const float*)d_in[0];
    const float* key   = (const float*)d_in[1];
    const float* value = (const float*)d_in[2];
    const float* Wq    = (const float*)d_in[3];
    const int*   qe    = (const int*)d_in[4];
    const int*   va    = (const int*)d_in[5];

    float* out  = (float*)d_out;
    float* qbuf = (float*)d_ws;

    const int out_n4 = NQ * DD / 4;  // 3,200,000 float4s
    zero_kernel<<<(out_n4 + 255) / 256, 256, 0, stream>>>(out, out_n4);

    dim3 ggrid(NQ / 16, 2);          // 3125 M-tiles x 2 (8 N-tiles per block)
    gemm_gelu_kernel<<<ggrid, 256, 0, stream>>>(query, Wq, qbuf);

    edge_kernel<<<NK / 4, 256, 0, stream>>>(qbuf, key, value, qe, va, out);
}